// DiffeomMap_2491081031960
// MI455X (gfx1250) — compile-verified
//
#include <hip/hip_runtime.h>

typedef float v2f __attribute__((ext_vector_type(2)));
typedef float v8f __attribute__((ext_vector_type(8)));

#define BATCH     16384
#define TILE_B    64
#define NWAVES    8
#define THREADS   256
#define MDIM      512   // num metabolites
#define HDIM      32
#define ODIM      16

// LDS layout (float offsets). Region0 holds X then (after X dies) EL.
// Region1 holds E1 then L1. Region2 holds E2.
#define OFF_R0    0                 // 64*512 floats (X / EL)
#define OFF_R1    (64*512)          // 64*256 floats (E1 / L1)
#define OFF_R2    (64*512 + 64*256) // 64*64  floats (E2)
#define SMEM_FLOATS (64*512 + 64*256 + 64*64)   // 53248 floats = 208 KB

__device__ __forceinline__ float relu_f(float v) { return v > 0.f ? v : 0.f; }

// dst[64][N] = relu(src[64][K] @ W[K][N] + bias[N])
// src/dst in LDS, W/bias in global (L2-resident).
// Fragment layout per CDNA5 ISA 7.12.2 for V_WMMA_F32_16X16X4_F32 (wave32):
//   A 16x4 : lane l<16 -> A[l][k0+0..1], lane l+16 -> A[l][k0+2..3]
//   B 4x16 : lane l<16 -> B[k0+0..1][l], lane l+16 -> B[k0+2..3][l]
//   C/D    : lane l, vgpr r -> D[r + 8*(l>=16)][l&15]
template <int K, int N>
__device__ void gemm_relu_layer(const float* __restrict__ W,
                                const float* __restrict__ bias,
                                const float* __restrict__ src,
                                float* __restrict__ dst,
                                int wave, int l16, int half)
{
    const v8f zero = {0.f, 0.f, 0.f, 0.f, 0.f, 0.f, 0.f, 0.f};
    for (int nt = wave; nt < N / 16; nt += NWAVES) {
        const int ncol = nt * 16 + l16;
        v8f acc[4];
        acc[0] = zero; acc[1] = zero; acc[2] = zero; acc[3] = zero;
        for (int k0 = 0; k0 < K; k0 += 4) {
            const int kb = k0 + 2 * half;
            v2f bf;
            bf.x = W[(size_t)(kb + 0) * N + ncol];
            bf.y = W[(size_t)(kb + 1) * N + ncol];
#pragma unroll
            for (int mt = 0; mt < 4; ++mt) {
                const float* ap = src + (mt * 16 + l16) * K + kb;
                v2f af;
                af.x = ap[0];
                af.y = ap[1];
                acc[mt] = __builtin_amdgcn_wmma_f32_16x16x4_f32(
                    false, af, false, bf, (short)0, acc[mt], false, false);
            }
        }
        const float bv = bias[ncol];
#pragma unroll
        for (int mt = 0; mt < 4; ++mt) {
#pragma unroll
            for (int r = 0; r < 8; ++r) {
                dst[(mt * 16 + r + 8 * half) * N + ncol] = relu_f(acc[mt][r] + bv);
            }
        }
    }
}

__global__ __launch_bounds__(THREADS)
void fused_metab_kernel(const float* __restrict__ x,
                        const float* __restrict__ eW1, const float* __restrict__ eb1,
                        const float* __restrict__ eW2, const float* __restrict__ eb2,
                        const float* __restrict__ lW1, const float* __restrict__ lb1,
                        const float* __restrict__ lW2, const float* __restrict__ lb2,
                        const float* __restrict__ dW1, const float* __restrict__ db1,
                        const float* __restrict__ dW2, const float* __restrict__ db2,
                        float* __restrict__ out)
{
    extern __shared__ float smem[];
    const int tid  = threadIdx.x;
    const int wave = tid >> 5;
    const int lane = tid & 31;
    const int l16  = lane & 15;
    const int half = lane >> 4;
    const int b0   = blockIdx.x * TILE_B;

    float* X  = smem + OFF_R0;   // [64][512]
    float* EL = smem + OFF_R0;   // [64][512]  (reuses X region after X is dead)
    float* E1 = smem + OFF_R1;   // [64][256]
    float* L1 = smem + OFF_R1;   // [64][256]  (reuses E1 region)
    float* E2 = smem + OFF_R2;   // [64][64]

    // ---- stage 0: load x tile [64,512] into LDS (float4 copies) ----
    {
        const float4* gx = (const float4*)(x + (size_t)b0 * 512);
        float4* sx = (float4*)X;
        for (int i = tid; i < TILE_B * 512 / 4; i += THREADS) sx[i] = gx[i];
    }
    __syncthreads();

    // ---- encoder + lift, all fp32 WMMA ----
    gemm_relu_layer<512, 256>(eW1, eb1, X,  E1, wave, l16, half);
    __syncthreads();
    gemm_relu_layer<256, 64>(eW2, eb2, E1, E2, wave, l16, half);
    __syncthreads();
    gemm_relu_layer<64,  256>(lW1, lb1, E2, L1, wave, l16, half);
    __syncthreads();
    gemm_relu_layer<256, 512>(lW2, lb2, L1, EL, wave, l16, half);
    __syncthreads();

    // ---- per-metabolite deconv: out[b,m,:] = relu(h[b,:] @ dW2[m] + db2[m]),
    //      h[b,j] = relu(el[b,m]*dW1[m,j] + db1[m,j]), computed on the fly as
    //      the A-fragment of a 16x16x4 fp32 WMMA (K=32, N=16). ----
    const v8f zero = {0.f, 0.f, 0.f, 0.f, 0.f, 0.f, 0.f, 0.f};
    for (int m = wave; m < MDIM; m += NWAVES) {
        // Preload the full B panel dW2[m] (32x16) as 8 K-step fragments.
        v2f bfr[8];
#pragma unroll
        for (int ks = 0; ks < 8; ++ks) {
            const int kb = ks * 4 + 2 * half;
            bfr[ks].x = dW2[((size_t)m * HDIM + kb + 0) * ODIM + l16];
            bfr[ks].y = dW2[((size_t)m * HDIM + kb + 1) * ODIM + l16];
        }
        const float ob = db2[m * ODIM + l16];
#pragma unroll
        for (int mt = 0; mt < 4; ++mt) {
            const float elv = EL[(mt * 16 + l16) * MDIM + m];
            v8f acc = zero;
#pragma unroll
            for (int ks = 0; ks < 8; ++ks) {
                const int kb = ks * 4 + 2 * half;
                v2f af;
                af.x = relu_f(elv * dW1[m * HDIM + kb + 0] + db1[m * HDIM + kb + 0]);
                af.y = relu_f(elv * dW1[m * HDIM + kb + 1] + db1[m * HDIM + kb + 1]);
                acc = __builtin_amdgcn_wmma_f32_16x16x4_f32(
                    false, af, false, bfr[ks], (short)0, acc, false, false);
            }
#pragma unroll
            for (int r = 0; r < 8; ++r) {
                const int brow = b0 + mt * 16 + r + 8 * half;
                out[((size_t)brow * MDIM + m) * ODIM + l16] = relu_f(acc[r] + ob);
            }
        }
    }
}

extern "C" void kernel_launch(void* const* d_in, const int* in_sizes, int n_in,
                              void* d_out, int out_size, void* d_ws, size_t ws_size,
                              hipStream_t stream) {
    (void)in_sizes; (void)n_in; (void)out_size; (void)d_ws; (void)ws_size;
    const float* x   = (const float*)d_in[0];
    const float* eW1 = (const float*)d_in[1];
    const float* eb1 = (const float*)d_in[2];
    const float* eW2 = (const float*)d_in[3];
    const float* eb2 = (const float*)d_in[4];
    const float* lW1 = (const float*)d_in[5];
    const float* lb1 = (const float*)d_in[6];
    const float* lW2 = (const float*)d_in[7];
    const float* lb2 = (const float*)d_in[8];
    const float* dW1 = (const float*)d_in[9];
    const float* db1 = (const float*)d_in[10];
    const float* dW2 = (const float*)d_in[11];
    const float* db2 = (const float*)d_in[12];
    float* out = (float*)d_out;

    const size_t smem_bytes = SMEM_FLOATS * sizeof(float);  // 208 KB (<320 KB WGP LDS)
    (void)hipFuncSetAttribute((const void*)fused_metab_kernel,
                              hipFuncAttributeMaxDynamicSharedMemorySize,
                              (int)smem_bytes);

    fused_metab_kernel<<<BATCH / TILE_B, THREADS, smem_bytes, stream>>>(
        x, eW1, eb1, eW2, eb2, lW1, lb1, lW2, lb2, dW1, db1, dW2, db2, out);
}